// GREEN_62869731278967
// MI455X (gfx1250) — compile-verified
//
#include <hip/hip_runtime.h>

// ---------------------------------------------------------------------------
// Zero-phase Butterworth (order 5) filtfilt over 4096 rows x 8192 fp32.
//
// MI455X reasoning:
//   * 256 MB min traffic / 23.3 TB/s => ~11 us floor; ~1.5 GFLOP total =>
//     strictly memory/latency bound. WMMA-izing the IIR (truncated FIR) would
//     inflate FLOPs 100-500x -> worse than the memory bound; skip WMMA.
//   * Serial-scan bottleneck broken by segment decomposition: max pole mag
//     0.8205 => 0.8205^128 ~ 1e-11 << fp32 eps, so a 128-sample warm-up
//     run-in makes segments independent to fp32 accuracy (exact init at the
//     true odd-extension row edges).
//   * CDNA5 data path: Tensor Data Mover (tensor_load_to_lds) stages each
//     256-row x 32-col tile with ONE instruction per workgroup chunk
//     (TENSORcnt pipelined, double buffered). TDM pad_enable (32 DWORDs ->
//     +4 DWORDs) reproduces the bank-conflict-free 36-float LDS row stride
//     in hardware. Fallback: per-lane global_load_async_to_lds_b128.
// ---------------------------------------------------------------------------

#define NLEN   8192          // row length
#define NROWS  4096
#define PADL   18            // filtfilt odd-extension pad
#define SEG    256           // output samples per segment
#define NSEG   (NLEN / SEG)  // 32
#define WUP    128           // IIR warm-up run-in (transient < 1e-11)
#define TR     256           // threads per block == rows per block
#define XS     36            // LDS x-tile row stride (floats) == TDM 32+4 pad
#define YS     33            // LDS y-tile row stride (floats; conflict-free)
#define SROW   8224          // scratch row stride (NLEN + 32, covers ext tail)

typedef unsigned int u32x4 __attribute__((ext_vector_type(4)));
typedef int          i32x4 __attribute__((ext_vector_type(4)));
typedef int          i32x8 __attribute__((ext_vector_type(8)));

// ---- feature detection ----------------------------------------------------
#if defined(__has_builtin)
#  if __has_builtin(__builtin_amdgcn_tensor_load_to_lds)
#    define HAVE_TDM 1
#  else
#    define HAVE_TDM 0
#  endif
#  if __has_builtin(__builtin_amdgcn_s_wait_tensorcnt)
#    define WAIT_TENSOR(n) __builtin_amdgcn_s_wait_tensorcnt(n)
#  endif
#  if __has_builtin(__builtin_amdgcn_s_wait_asynccnt)
#    define WAIT_ASYNC(n) __builtin_amdgcn_s_wait_asynccnt(n)
#  endif
#else
#  define HAVE_TDM 0
#endif
#ifndef WAIT_TENSOR
#  define WAIT_TENSOR(n) asm volatile("s_wait_tensorcnt %0" :: "i"(n) : "memory")
#endif
#ifndef WAIT_ASYNC
#  define WAIT_ASYNC(n) asm volatile("s_wait_asynccnt %0" :: "i"(n) : "memory")
#endif

__device__ __forceinline__ unsigned lds_off_of(const void* p) {
  // generic pointer to LDS: low 32 bits are the LDS byte offset.
  return (unsigned)(unsigned long long)p;
}

// ---- TDM staging: one descriptor moves the whole 256x32 fp32 tile ---------
#if HAVE_TDM
__device__ __forceinline__ void tdm_issue(const float* src, unsigned pitchElems,
                                          int rowBase, int colBase,
                                          unsigned ldsByte) {
  const unsigned long long ga =
      (unsigned long long)(const void*)src +
      ((unsigned long long)(unsigned)rowBase * pitchElems +
       (unsigned long long)(unsigned)colBase) * 4ull;

  u32x4 g0;
  g0[0] = 1u;                                   // count=1, user mode, no gather
  g0[1] = ldsByte;                              // lds_addr
  g0[2] = (unsigned)ga;                         // global_addr[31:0]
  g0[3] = (unsigned)((ga >> 32) & 0x01FFFFFFull) | (2u << 30);  // [56:32]|type=2

  // data_size=4B (2), pad_enable, pad_interval=4 (every 32 DWORDs),
  // pad_amount=3 (+4 DWORDs) -> LDS row stride 36 floats == XS.
  i32x8 g1;
  g1[0] = (2 << 16) | (1 << 20) | (4 << 22) | (3 << 25);
  g1[1] = 0;                                    // tensor_dim0 lo16 (2^30) -> 0
  g1[2] = 0x4000;                               // tensor_dim0 hi16, dim1 lo16=0
  g1[3] = 0x4000 | (32 << 16);                  // tensor_dim1 hi16 | tile_dim0=32
  g1[4] = TR;                                   // tile_dim1=256, tile_dim2=0
  g1[5] = (int)pitchElems;                      // tensor_dim0_stride lo32
  g1[6] = 0;                                    // stride hi16 | dim1_stride lo16
  g1[7] = 0;                                    // dim1_stride hi32

  const i32x4 z4 = {0, 0, 0, 0};
#if __clang_major__ >= 23
  const i32x8 z8 = {0, 0, 0, 0, 0, 0, 0, 0};
  __builtin_amdgcn_tensor_load_to_lds(g0, g1, z4, z4, z8, 0);
#else
  __builtin_amdgcn_tensor_load_to_lds(g0, g1, z4, z4, 0);
#endif
}
#else
// Fallback: per-lane async global->LDS b128 (8 lanes cover one row's 32 cols)
__device__ __forceinline__ void async_load16(unsigned lds, const float* g) {
  asm volatile("global_load_async_to_lds_b128 %0, %1, off"
               :: "v"(lds), "v"(g) : "memory");
}
#endif

__device__ __forceinline__ void stage_chunk(const float* __restrict__ src,
                                            unsigned pitchElems, int rowBase,
                                            int colBase, unsigned ldsBase,
                                            int tid) {
#if HAVE_TDM
  if (tid < 32) tdm_issue(src, pitchElems, rowBase, colBase, ldsBase);
#else
#pragma unroll
  for (int j = 0; j < 8; ++j) {
    const int fid = j * TR + tid;       // 0..2047 float4 ids
    const int r   = fid >> 3;           // 0..255
    const int c4  = (fid & 7) << 2;     // 0,4,...,28
    async_load16(ldsBase + (unsigned)(r * XS + c4) * 4u,
                 src + (size_t)(rowBase + r) * pitchElems + (size_t)(colBase + c4));
  }
#endif
}

__device__ __forceinline__ void stage_wait_one(int tid) {
#if HAVE_TDM
  if (tid < 32) WAIT_TENSOR(1);   // previous tile done; next stays in flight
#else
  (void)tid; WAIT_ASYNC(8);
#endif
}

__device__ __forceinline__ void stage_wait_zero(int tid) {
#if HAVE_TDM
  if (tid < 32) WAIT_TENSOR(0);
#else
  (void)tid; WAIT_ASYNC(0);
#endif
}

// ---------------------------------------------------------------------------
// Forward pass: y = lfilter(b, a, odd_ext(x), zi*ext[first]) per segment,
// with WUP-sample run-in; stores y[s..e) (+ ext tail for last segment) to ws.
// ---------------------------------------------------------------------------
__global__ __launch_bounds__(TR) void GREEN_fwd_iir(
    const float* __restrict__ x, const float* __restrict__ bc,
    const float* __restrict__ ac, const float* __restrict__ zc,
    float* __restrict__ ws) {
  __shared__ float xb[2][TR * XS];
  __shared__ float yt[TR * YS];

  const int tid = threadIdx.x;
  const int rowBase = blockIdx.x * TR;
  const int seg = blockIdx.y;
  const int row = rowBase + tid;

  const float b0 = bc[0], b1 = bc[1], b2 = bc[2], b3 = bc[3], b4 = bc[4], b5 = bc[5];
  const float a1 = ac[1], a2 = ac[2], a3 = ac[3], a4 = ac[4], a5 = ac[5];
  const float q0 = zc[0], q1 = zc[1], q2 = zc[2], q3 = zc[3], q4 = zc[4];

  const int s = seg * SEG;
  const int e = s + SEG;
  const int fwdStart = (seg == 0) ? -PADL : (s - WUP);        // ext coords
  const int fwdEnd   = (seg == NSEG - 1) ? (NLEN + PADL) : e;
  const int c0 = (seg == 0) ? 0 : (s - WUP);                  // staged col range
  const int c1 = (seg == NSEG - 1) ? NLEN : e;
  const int nCh = (c1 - c0) >> 5;

  const float x0 = x[(size_t)row * NLEN];
  const float xl = x[(size_t)row * NLEN + (NLEN - 1)];

  const unsigned ldsB0 = lds_off_of(&xb[0][0]);
  const unsigned ldsB1 = lds_off_of(&xb[1][0]);

  float z0 = 0.f, z1 = 0.f, z2 = 0.f, z3 = 0.f, z4 = 0.f;
  int t = fwdStart;

  stage_chunk(x, NLEN, rowBase, c0, ldsB0, tid);  // prefetch chunk 0

  for (int k = 0; k < nCh; ++k) {
    const int base = c0 + (k << 5);
    if (k + 1 < nCh) {
      stage_chunk(x, NLEN, rowBase, base + 32, ((k + 1) & 1) ? ldsB1 : ldsB0, tid);
      stage_wait_one(tid);
    } else {
      stage_wait_zero(tid);
    }
    __syncthreads();   // issuing wave(s) waited -> tile visible to the block

    const float* xr = &xb[k & 1][tid * XS];
    while (t < fwdEnd) {
      int c;
      if (t < 0)          c = -t;                 // left odd-extension source
      else if (t < NLEN)  c = t;
      else                c = 2 * NLEN - 2 - t;   // right odd-extension source
      if (c < base || c >= base + 32) break;
      const float xv = xr[c - base];              // ds_load_b32, conflict-free
      const float xt = (t < 0) ? (2.0f * x0 - xv)
                               : ((t >= NLEN) ? (2.0f * xl - xv) : xv);
      if (t == fwdStart) {                        // zi * first-input init
        z0 = q0 * xt; z1 = q1 * xt; z2 = q2 * xt; z3 = q3 * xt; z4 = q4 * xt;
      }
      const float y = fmaf(b0, xt, z0);           // DF2T order-5 step
      z0 = fmaf(b1, xt, z1) - a1 * y;
      z1 = fmaf(b2, xt, z2) - a2 * y;
      z2 = fmaf(b3, xt, z3) - a3 * y;
      z3 = fmaf(b4, xt, z4) - a4 * y;
      z4 = b5 * xt - a5 * y;
      if (t >= s) {
        if (t < NLEN) yt[tid * YS + (t - base)] = y;       // tile for writeback
        else          ws[(size_t)row * SROW + t] = y;      // tiny ext tail
      }
      ++t;
    }

    if (base >= s) {   // chunk fully inside the stored range: coalesced flush
      __syncthreads();
#pragma unroll
      for (int j = 0; j < 32; ++j) {
        const int eid = j * TR + tid;
        const int r = eid >> 5, cc = eid & 31;   // wave writes one row's 128B
        ws[(size_t)(rowBase + r) * SROW + (size_t)(base + cc)] = yt[r * YS + cc];
      }
    }
    __syncthreads();   // tiles free before next stage overwrites them
  }
}

// ---------------------------------------------------------------------------
// Backward pass: filter ws rows in reverse with WUP run-in from above,
// exact init at the ext top for the last segment; writes out[s..e).
// ---------------------------------------------------------------------------
__global__ __launch_bounds__(TR) void GREEN_bwd_iir(
    const float* __restrict__ ws, const float* __restrict__ bc,
    const float* __restrict__ ac, const float* __restrict__ zc,
    float* __restrict__ out) {
  __shared__ float xb[2][TR * XS];
  __shared__ float yt[TR * YS];

  const int tid = threadIdx.x;
  const int rowBase = blockIdx.x * TR;
  const int seg = blockIdx.y;

  const float b0 = bc[0], b1 = bc[1], b2 = bc[2], b3 = bc[3], b4 = bc[4], b5 = bc[5];
  const float a1 = ac[1], a2 = ac[2], a3 = ac[3], a4 = ac[4], a5 = ac[5];
  const float q0 = zc[0], q1 = zc[1], q2 = zc[2], q3 = zc[3], q4 = zc[4];

  const int s = seg * SEG;
  const int e = s + SEG;
  const int topT = (seg == NSEG - 1) ? (NLEN + PADL - 1) : (e + WUP - 1);
  const int nCh = ((topT - s) + 32) >> 5;   // ceil((topT+1-s)/32)

  const unsigned ldsB0 = lds_off_of(&xb[0][0]);
  const unsigned ldsB1 = lds_off_of(&xb[1][0]);

  float z0 = 0.f, z1 = 0.f, z2 = 0.f, z3 = 0.f, z4 = 0.f;
  int t = topT;

  stage_chunk(ws, SROW, rowBase, s + ((nCh - 1) << 5),
              ((nCh - 1) & 1) ? ldsB1 : ldsB0, tid);  // prefetch top chunk

  for (int k = nCh - 1; k >= 0; --k) {
    const int base = s + (k << 5);
    if (k > 0) {
      stage_chunk(ws, SROW, rowBase, base - 32, ((k - 1) & 1) ? ldsB1 : ldsB0, tid);
      stage_wait_one(tid);
    } else {
      stage_wait_zero(tid);
    }
    __syncthreads();

    const float* yr = &xb[k & 1][tid * XS];
    while (t >= s) {
      if (t < base) break;
      const float yv = yr[t - base];
      if (t == topT) {   // zi * first reversed input (exact at ext top)
        z0 = q0 * yv; z1 = q1 * yv; z2 = q2 * yv; z3 = q3 * yv; z4 = q4 * yv;
      }
      const float o = fmaf(b0, yv, z0);
      z0 = fmaf(b1, yv, z1) - a1 * o;
      z1 = fmaf(b2, yv, z2) - a2 * o;
      z2 = fmaf(b3, yv, z3) - a3 * o;
      z3 = fmaf(b4, yv, z4) - a4 * o;
      z4 = b5 * yv - a5 * o;
      if (t < e) yt[tid * YS + (t - base)] = o;
      --t;
    }

    if (base + 32 <= e) {  // chunk fully inside output range: coalesced flush
      __syncthreads();
#pragma unroll
      for (int j = 0; j < 32; ++j) {
        const int eid = j * TR + tid;
        const int r = eid >> 5, cc = eid & 31;
        out[(size_t)(rowBase + r) * NLEN + (size_t)(base + cc)] = yt[r * YS + cc];
      }
    }
    __syncthreads();
  }
}

// ---------------------------------------------------------------------------
extern "C" void kernel_launch(void* const* d_in, const int* in_sizes, int n_in,
                              void* d_out, int out_size, void* d_ws, size_t ws_size,
                              hipStream_t stream) {
  (void)in_sizes; (void)n_in; (void)out_size; (void)ws_size;
  const float* x  = (const float*)d_in[0];
  const float* b  = (const float*)d_in[1];
  const float* a  = (const float*)d_in[2];
  const float* zi = (const float*)d_in[3];
  float* ws  = (float*)d_ws;   // NROWS * SROW floats of forward-pass output
  float* out = (float*)d_out;

  dim3 grid(NROWS / TR, NSEG);  // (16, 32) blocks, 8 wave32 waves each
  GREEN_fwd_iir<<<grid, TR, 0, stream>>>(x, b, a, zi, ws);
  GREEN_bwd_iir<<<grid, TR, 0, stream>>>(ws, b, a, zi, out);
}